// CPUWaveletTransform_44659069944424
// MI455X (gfx1250) — compile-verified
//
#include <hip/hip_runtime.h>

typedef __attribute__((ext_vector_type(2))) float v2f;
typedef __attribute__((ext_vector_type(8))) float v8f;

#define NCHUNK 10          // K=40 covered as 10 x K=4 WMMA steps (band needs 38)
#define MT_PER_WAVE 4      // m-tiles per wave (amortizes A-fragment loads)
#define WAVES_PER_BLOCK 8  // 256 threads, wave32
#define FEAT 128
#define FTILES (FEAT / 16)

// pywt db4 (true convolution): y[m] = sum_j F[j] * xp[2m+j], F[j] = DEC[7-j]
constexpr float FLO_C[8] = { 0.23037781330885523f,  0.7148465705525415f,
                             0.6308807679295904f,  -0.02798376941698385f,
                            -0.18703481171888114f,  0.030841381835986965f,
                             0.032883011666982945f,-0.010597401784997278f };
constexpr float FHI_C[8] = {-0.010597401784997278f,-0.032883011666982945f,
                             0.030841381835986965f, 0.18703481171888114f,
                            -0.02798376941698385f, -0.6308807679295904f,
                             0.7148465705525415f,  -0.23037781330885523f };

// Banded A-fragment table, laid out so a lane's (v=0,v=1) pair is one
// aligned 8-byte load: w[chunk][kh][mr][v] with W[mr,tloc]=F[tloc-2*mr],
// tloc = 4*chunk + 2*kh + v.  ~5 KB total, compile-time initialized.
struct alignas(16) WTabs {
    float lo[NCHUNK][2][16][2];
    float hi[NCHUNK][2][16][2];
};

constexpr WTabs make_wtabs() {
    WTabs t{};
    for (int c = 0; c < NCHUNK; ++c)
        for (int kh = 0; kh < 2; ++kh)
            for (int mr = 0; mr < 16; ++mr)
                for (int v = 0; v < 2; ++v) {
                    int j = 4 * c + 2 * kh + v - 2 * mr;
                    t.lo[c][kh][mr][v] = (j >= 0 && j < 8) ? FLO_C[j] : 0.0f;
                    t.hi[c][kh][mr][v] = (j >= 0 && j < 8) ? FHI_C[j] : 0.0f;
                }
    return t;
}

__device__ __constant__ WTabs c_wt = make_wtabs();

// Half-sample symmetric extension index map (pad 7, drop first sample):
// xp[t] = x[5-t] (t<6), x[t-6] (6<=t<=T+5), x[2T+5-t] (t>T+5); clamped safe.
__device__ __forceinline__ int map_sym(int t, int T) {
    int idx = t - 6;
    idx = (idx < 0)  ? (-1 - idx)        : idx;   // left reflect
    idx = (idx >= T) ? (2 * T - 1 - idx) : idx;   // right reflect
    idx = idx < 0 ? 0 : idx;                      // safety clamp (zero-weight rows)
    idx = idx > T - 1 ? T - 1 : idx;
    return idx;
}

// One DWT level. in:(B,T,FEAT) -> outA,outD:(B,Tout,FEAT).
// Each wave computes MT_PER_WAVE tiles of 16 output positions x 16 features
// with v_wmma_f32_16x16x4_f32 against banded filter matrices held in VGPRs.
__global__ __launch_bounds__(256) void dwt_level_wmma(
    const float* __restrict__ in, float* __restrict__ outA,
    float* __restrict__ outD, int T, int Tout, int nMTiles, int nChunks,
    int totalWaves)
{
    const int lane = threadIdx.x & 31;
    const int w = blockIdx.x * WAVES_PER_BLOCK + (threadIdx.x >> 5);
    if (w >= totalWaves) return;  // wave-uniform exit: survivors keep EXEC all-1s

    const int ftile = w % FTILES;
    const int chunk = (w / FTILES) % nChunks;
    const int b     = w / (FTILES * nChunks);

    const int mr = lane & 15;  // A: M row / B,D: N column
    const int kh = lane >> 4;  // lane half selects K (A/B) and M+8 (C/D)

    // Load A fragments once from the constant table (one b64 load each).
    v2f alo[NCHUNK], ahi[NCHUNK];
#pragma unroll
    for (int c = 0; c < NCHUNK; ++c) {
        alo[c] = *(const v2f*)(&c_wt.lo[c][kh][mr][0]);
        ahi[c] = *(const v2f*)(&c_wt.hi[c][kh][mr][0]);
    }

    const int fcol = ftile * 16 + mr;
    const float* basein = in + (size_t)b * T * FEAT + fcol;

    const int mtile0 = chunk * MT_PER_WAVE;
    for (int i = 0; i < MT_PER_WAVE; ++i) {
        const int mtile = mtile0 + i;
        if (mtile >= nMTiles) break;  // wave-uniform
        const int m0 = mtile * 16;
        const int t0 = 2 * m0;

        v8f accA = {};
        v8f accD = {};

        // Interior tiles (all but first/last per signal): no reflection, so
        // B-fragment addresses are base + compile-time-constant offsets that
        // fold into the load immediates.  Wave-uniform branch.
        if (t0 >= 6 && t0 + 4 * NCHUNK - 1 <= T + 5) {
            const float* p = basein + (size_t)(t0 - 6 + 2 * kh) * FEAT;
#pragma unroll
            for (int c = 0; c < NCHUNK; ++c) {
                v2f bfrag;
                bfrag[0] = p[(size_t)(4 * c) * FEAT];
                bfrag[1] = p[(size_t)(4 * c + 1) * FEAT];
                accA = __builtin_amdgcn_wmma_f32_16x16x4_f32(
                    false, alo[c], false, bfrag, (short)0, accA, false, false);
                accD = __builtin_amdgcn_wmma_f32_16x16x4_f32(
                    false, ahi[c], false, bfrag, (short)0, accD, false, false);
            }
        } else {
#pragma unroll
            for (int c = 0; c < NCHUNK; ++c) {
                v2f bfrag;
#pragma unroll
                for (int v = 0; v < 2; ++v) {
                    const int t = t0 + 4 * c + 2 * kh + v;
                    bfrag[v] = basein[(size_t)map_sym(t, T) * FEAT];
                }
                accA = __builtin_amdgcn_wmma_f32_16x16x4_f32(
                    false, alo[c], false, bfrag, (short)0, accA, false, false);
                accD = __builtin_amdgcn_wmma_f32_16x16x4_f32(
                    false, ahi[c], false, bfrag, (short)0, accD, false, false);
            }
        }

        const int mbase = m0 + kh * 8;
        float* pa = outA + ((size_t)b * Tout + mbase) * FEAT + fcol;
        float* pd = outD + ((size_t)b * Tout + mbase) * FEAT + fcol;
        if (m0 + 16 <= Tout) {  // full tile: unconditional stores
#pragma unroll
            for (int r = 0; r < 8; ++r) {
                pa[(size_t)r * FEAT] = accA[r];
                pd[(size_t)r * FEAT] = accD[r];
            }
        } else {                // final partial tile: per-row predication
#pragma unroll
            for (int r = 0; r < 8; ++r) {
                if (mbase + r < Tout) {
                    pa[(size_t)r * FEAT] = accA[r];
                    pd[(size_t)r * FEAT] = accD[r];
                }
            }
        }
    }
}

static void launch_level(const float* in, float* outA, float* outD, int B,
                         int T, hipStream_t s) {
    const int Tout = (T + 7) / 2;
    const int nMTiles = (Tout + 15) / 16;
    const int nChunks = (nMTiles + MT_PER_WAVE - 1) / MT_PER_WAVE;
    const int totalWaves = B * FTILES * nChunks;
    const int blocks = (totalWaves + WAVES_PER_BLOCK - 1) / WAVES_PER_BLOCK;
    dwt_level_wmma<<<blocks, 256, 0, s>>>(in, outA, outD, T, Tout, nMTiles,
                                          nChunks, totalWaves);
}

extern "C" void kernel_launch(void* const* d_in, const int* in_sizes, int n_in,
                              void* d_out, int out_size, void* d_ws,
                              size_t ws_size, hipStream_t stream) {
    (void)in_sizes; (void)n_in; (void)out_size; (void)ws_size;
    const float* x = (const float*)d_in[0];
    float* out = (float*)d_out;

    const int B = 64, T0 = 4096;
    const int T1 = 2051, T2 = 1029, T3 = 518;  // per-level output lengths

    const size_t szL1 = (size_t)B * T1 * FEAT;
    const size_t szL2 = (size_t)B * T2 * FEAT;
    const size_t szL3 = (size_t)B * T3 * FEAT;

    // Output layout: [low(518) | cD3(518) | cD2(1029) | cD1(2051)]
    float* low = out;
    float* cd3 = out + szL3;
    float* cd2 = out + 2 * szL3;
    float* cd1 = out + 2 * szL3 + szL2;

    // Workspace: cA1 then cA2 (~101 MB)
    float* ca1 = (float*)d_ws;
    float* ca2 = ca1 + szL1;

    launch_level(x,   ca1, cd1, B, T0, stream);  // level 1
    launch_level(ca1, ca2, cd2, B, T1, stream);  // level 2
    launch_level(ca2, low, cd3, B, T2, stream);  // level 3
}